// FieldOfJunctions_78030965834175
// MI455X (gfx1250) — compile-verified
//
#include <hip/hip_runtime.h>
#include <math.h>

// ---------------------------------------------------------------------------
// Field of Junctions on gfx1250 (MI455X).
// Pipeline:
//   K1  reduce_wmma      : [3x16]x[16x9216] channel reduction via
//                          V_WMMA_F32_16X16X4_F32 (4 chained K-steps,
//                          branchless operand staging -> no exec juggling).
//   K2  bn_silu3         : per-channel batchnorm + silu (single block).
//   K3  zero params
//   K4.. 8x descent_round: wave32 = 31 candidates per patch position,
//                          closed-form loss (18 accumulators), shfl argmin.
//   K5  derive           : per-position derived geometry (14 floats).
//   K6  final            : gather-fold of lb / count, 1x1 conv, BN, silu.
// Deterministic: no atomics, fixed reduction trees, all state re-derived
// from d_in each call.
// ---------------------------------------------------------------------------

namespace {
constexpr int Himg = 96, Wimg = 96, Ppat = 8, HP = 89, WP = 89;
constexpr int NPOS = HP * WP;      // 7921
constexpr int NPIX = Himg * Wimg;  // 9216
constexpr int NV = 31;
constexpr float PI_F      = 3.14159265358979323846f;
constexpr float TWO_PI_F  = 6.28318530717958647692f;
constexpr float INV_PI_F  = 0.31830988618379067154f;
constexpr float TAU_F     = 0.1f;
constexpr float INV_ETA   = 100.0f;  // 1/0.01
constexpr float INV_DELTA = 20.0f;   // 1/0.05
}

typedef float v2f __attribute__((ext_vector_type(2)));
typedef float v8f __attribute__((ext_vector_type(8)));

// jnp.mod(x, 2pi): result in [0, 2pi)
__device__ __forceinline__ float pmod2pi(float x) {
  float m = fmodf(x, TWO_PI_F);
  return (m < 0.f) ? m + TWO_PI_F : m;
}

__device__ __forceinline__ float pow35(float x) {
  float x2 = x * x, x4 = x2 * x2, x8 = x4 * x4, x16 = x8 * x8, x32 = x16 * x16;
  return x32 * x2 * x;  // x^35 (odd power, sign preserved)
}

// linspace(-1, 1, 8)
__device__ __forceinline__ float grid8(int i) {
  return -1.f + (2.f / 7.f) * (float)i;
}
// linspace(0, 2pi, 32)[:31]
__device__ __forceinline__ float angle_range(int n) {
  return TWO_PI_F * (float)n / 31.f;
}
// linspace(-3, 3, 31)
__device__ __forceinline__ float xy_range(int n) {
  return -3.f + 0.2f * (float)n;
}

struct Deriv {
  float sn0, cs0, sn1, cs1, sn2, cs2, sn3, cs3;  // sincos of a1,a2,a3,a4
  float s13, s42, g13, g42, x0, y0;
};

__device__ __forceinline__ Deriv make_deriv(const float c5[5]) {
  Deriv D;
  float a0 = pmod2pi(c5[0]), a1 = pmod2pi(c5[1]), a2 = pmod2pi(c5[2]);
  float t;
  if (a0 > a1) { t = a0; a0 = a1; a1 = t; }
  if (a1 > a2) { t = a1; a1 = a2; a2 = t; }
  if (a0 > a1) { t = a0; a0 = a1; a1 = t; }
  // sorted: ref a1=a0, a2=a1, a3=a2
  float a4 = 0.5f * (a0 + a2) + ((pmod2pi(0.5f * (a0 - a2)) >= PI_F) ? PI_F : 0.f);
  float m42 = pmod2pi(a1 - a4);
  float m13 = pmod2pi(a2 - a0);
  D.s42 = (m42 < PI_F) ? 1.f : -1.f;
  D.s13 = (m13 < PI_F) ? 1.f : -1.f;
  D.g42 = pow35(m42 * INV_PI_F - 1.f) * TAU_F;
  D.g13 = pow35(m13 * INV_PI_F - 1.f) * TAU_F;
  sincosf(a0, &D.sn0, &D.cs0);
  sincosf(a1, &D.sn1, &D.cs1);
  sincosf(a2, &D.sn2, &D.cs2);
  sincosf(a4, &D.sn3, &D.cs3);
  D.x0 = c5[3]; D.y0 = c5[4];
  return D;
}

__device__ __forceinline__ void dist_pix(const Deriv& D, float lx, float ly,
                                         float& d13, float& d42) {
  float l1 = D.cs0 * ly - D.sn0 * lx;
  float l2 = D.cs1 * ly - D.sn1 * lx;
  float l3 = D.cs2 * ly - D.sn2 * lx;
  float l4 = D.cs3 * ly - D.sn3 * lx;
  d42 = D.s42 * fminf(D.s42 * l4, -D.s42 * l2) + D.g42;
  d13 = D.s13 * fminf(D.s13 * l1, -D.s13 * l3) + D.g13;
}

// Loss up to a candidate-invariant affine transform (argmin-equivalent):
// loss' = sum_c [ -2 sum_l mu_cl num_cl + sum_lm mu_cl mu_cm W_lm ]
__device__ float eval_loss(const float* __restrict__ xr, const float c5[5],
                           int h, int w) {
  Deriv D = make_deriv(c5);
  float den0 = 0.f, den1 = 0.f, den2 = 0.f;
  float num[3][3] = {{0.f}};
  float ww00 = 0.f, ww01 = 0.f, ww02 = 0.f, ww11 = 0.f, ww12 = 0.f, ww22 = 0.f;
  for (int p = 0; p < Ppat; ++p) {
    float ly = grid8(p) - D.y0;
    const float* r0 = xr + (h + p) * Wimg + w;
    const float* r1 = r0 + NPIX;
    const float* r2 = r1 + NPIX;
#pragma unroll
    for (int k = 0; k < Ppat; ++k) {
      float lx = grid8(k) - D.x0;
      float d13, d42;
      dist_pix(D, lx, ly, d13, d42);
      float h1 = 0.5f + INV_PI_F * atanf(d13 * INV_ETA);
      float h2 = 0.5f + INV_PI_F * atanf(d42 * INV_ETA);
      float w0 = 1.f - h1, w1 = h1 * (1.f - h2), w2 = h1 * h2;
      den0 += w0; den1 += w1; den2 += w2;
      ww00 += w0 * w0; ww01 += w0 * w1; ww02 += w0 * w2;
      ww11 += w1 * w1; ww12 += w1 * w2; ww22 += w2 * w2;
      float i0 = r0[k], i1 = r1[k], i2 = r2[k];
      num[0][0] += i0 * w0; num[0][1] += i0 * w1; num[0][2] += i0 * w2;
      num[1][0] += i1 * w0; num[1][1] += i1 * w1; num[1][2] += i1 * w2;
      num[2][0] += i2 * w0; num[2][1] += i2 * w1; num[2][2] += i2 * w2;
    }
  }
  float id0 = 1.f / (den0 + 1e-10f);
  float id1 = 1.f / (den1 + 1e-10f);
  float id2 = 1.f / (den2 + 1e-10f);
  float loss = 0.f;
#pragma unroll
  for (int c = 0; c < 3; ++c) {
    float m0 = num[c][0] * id0, m1 = num[c][1] * id1, m2 = num[c][2] * id2;
    loss += -2.f * (m0 * num[c][0] + m1 * num[c][1] + m2 * num[c][2])
          + m0 * m0 * ww00 + m1 * m1 * ww11 + m2 * m2 * ww22
          + 2.f * (m0 * m1 * ww01 + m0 * m2 * ww02 + m1 * m2 * ww12);
  }
  return loss;
}

// ---------------------------------------------------------------------------
// K1: y[o, pix] = sum_c W[o,c] * x[c, pix] via V_WMMA_F32_16X16X4_F32.
// A = W padded to 16x16 (rows 3..15 zero), K=16 -> 4 chained WMMAs.
// A layout (32-bit, 16x4): lanes 0-15 hold M=lane, {v0=K0, v1=K1};
//                          lanes 16-31 hold M=lane-16, {v0=K2, v1=K3}.
// B layout (4x16): v0 = row (2*hi), v1 = row (2*hi+1), N = lane&15.
// D layout: vgpr r -> lanes 0-15: (M=r, N=lane), lanes 16-31: (M=r+8).
// Grid = exactly 576 full waves: EXEC all-1s at every WMMA.
// All operands are staged branchlessly (clamped index + 0/1 mask -> cndmask,
// no exec save/restore) before the WMMA chain, so the 4 WMMAs issue
// back-to-back behind a single load wait.
// ---------------------------------------------------------------------------
__global__ void reduce_wmma_kernel(const float* __restrict__ x,
                                   const float* __restrict__ w16,
                                   float* __restrict__ y) {
  int wave = (int)((blockIdx.x * blockDim.x + threadIdx.x) >> 5);
  int lane = (int)(threadIdx.x & 31);
  int col = lane & 15;
  int hi = lane >> 4;
  int n0 = wave * 16;
  int colc = (col < 3) ? col : 0;      // clamped row (branchless pad)
  float msk = (col < 3) ? 1.f : 0.f;   // zero rows 3..15 of A

  v2f a[4], b[4];
#pragma unroll
  for (int q = 0; q < 4; ++q) {
    int kbase = 4 * q + 2 * hi;
    a[q].x = w16[colc * 16 + kbase] * msk;
    a[q].y = w16[colc * 16 + kbase + 1] * msk;
    b[q].x = x[(size_t)(kbase)     * NPIX + n0 + col];
    b[q].y = x[(size_t)(kbase + 1) * NPIX + n0 + col];
  }

  v8f acc = {};
#pragma unroll
  for (int q = 0; q < 4; ++q) {
    acc = __builtin_amdgcn_wmma_f32_16x16x4_f32(
        /*neg_a=*/false, a[q], /*neg_b=*/false, b[q],
        /*c_mod=*/(short)0, acc, /*reuse_a=*/false, /*reuse_b=*/false);
  }

  if (hi == 0) {
    y[0 * NPIX + n0 + col] = acc[0];
    y[1 * NPIX + n0 + col] = acc[1];
    y[2 * NPIX + n0 + col] = acc[2];
  }
}

__device__ float block_reduce_sum(float v, float* sbuf) {
  int lane = threadIdx.x & 31, wid = threadIdx.x >> 5;
#pragma unroll
  for (int o = 16; o; o >>= 1) v += __shfl_xor(v, o, 32);
  if (lane == 0) sbuf[wid] = v;
  __syncthreads();
  int nw = (blockDim.x + 31) >> 5;
  if (wid == 0) {
    float r = (lane < nw) ? sbuf[lane] : 0.f;
#pragma unroll
    for (int o = 16; o; o >>= 1) r += __shfl_xor(r, o, 32);
    if (lane == 0) sbuf[0] = r;
  }
  __syncthreads();
  float r = sbuf[0];
  __syncthreads();
  return r;
}

// K2: per-channel BN + silu over y (3 x 9216), in place.
__global__ __launch_bounds__(1024) void bn_silu3_kernel(
    float* __restrict__ y, const float* __restrict__ gamma,
    const float* __restrict__ beta) {
  __shared__ float sbuf[32];
  for (int c = 0; c < 3; ++c) {
    float s = 0.f, ss = 0.f;
    for (int i = threadIdx.x; i < NPIX; i += blockDim.x) {
      float v = y[c * NPIX + i];
      s += v; ss += v * v;
    }
    float sum = block_reduce_sum(s, sbuf);
    float sqs = block_reduce_sum(ss, sbuf);
    float mean = sum / (float)NPIX;
    float var = sqs / (float)NPIX - mean * mean;
    float inv = rsqrtf(var + 1e-5f);
    float g = gamma[c], b = beta[c];
    for (int i = threadIdx.x; i < NPIX; i += blockDim.x) {
      float v = y[c * NPIX + i];
      float bn = g * (v - mean) * inv + b;
      y[c * NPIX + i] = bn / (1.f + expf(-bn));  // silu
    }
    __syncthreads();
  }
}

__global__ void zero_kernel(float* __restrict__ p, int n) {
  int i = blockIdx.x * blockDim.x + threadIdx.x;
  if (i < n) p[i] = 0.f;
}

// K4: one wave32 per patch position; lane = candidate (31 real, lane 31 INF).
__global__ void descent_kernel(const float* __restrict__ xr,
                               float* __restrict__ params, int round) {
  int gtid = blockIdx.x * blockDim.x + threadIdx.x;
  int wave = gtid >> 5;
  int lane = gtid & 31;
  if (wave >= NPOS) return;  // wave-uniform exit
  float pr[5];
#pragma unroll
  for (int i = 0; i < 5; ++i) pr[i] = params[i * NPOS + wave];
  int n = (lane < NV) ? lane : (NV - 1);
  float c5[5] = {pr[0], pr[1], pr[2], pr[3], pr[4]};
  if (round < 5) {
    float d = (round < 3) ? angle_range(n) : xy_range(n);
    c5[round] += d;
  } else {
    int j = round - 5;
    float t = xy_range(n);
    c5[3] = pr[3] + cosf(pr[j]) * t;
    c5[4] = pr[4] + sinf(pr[j]) * t;
  }
  int h = wave / WP, w = wave % WP;
  float loss = eval_loss(xr, c5, h, w);
  if (lane >= NV) loss = INFINITY;
  // wave32 argmin (lowest candidate wins ties, matching jnp.argmin)
  int best = n;
  float bl = loss;
#pragma unroll
  for (int o = 16; o; o >>= 1) {
    float ol = __shfl_xor(bl, o, 32);
    int oi = __shfl_xor(best, o, 32);
    if (ol < bl || (ol == bl && oi < best)) { bl = ol; best = oi; }
  }
  if (lane == best) {
    if (round < 5) {
      params[round * NPOS + wave] = c5[round];
    } else {
      params[3 * NPOS + wave] = c5[3];
      params[4 * NPOS + wave] = c5[4];
    }
  }
}

// K5: per-position derived geometry for the final gather (14 floats).
__global__ void derive_kernel(const float* __restrict__ params,
                              float* __restrict__ deriv) {
  int pos = blockIdx.x * blockDim.x + threadIdx.x;
  if (pos >= NPOS) return;
  float c5[5];
#pragma unroll
  for (int i = 0; i < 5; ++i) c5[i] = params[i * NPOS + pos];
  Deriv D = make_deriv(c5);
  deriv[0 * NPOS + pos] = D.sn0;  deriv[1 * NPOS + pos] = D.cs0;
  deriv[2 * NPOS + pos] = D.sn1;  deriv[3 * NPOS + pos] = D.cs1;
  deriv[4 * NPOS + pos] = D.sn2;  deriv[5 * NPOS + pos] = D.cs2;
  deriv[6 * NPOS + pos] = D.sn3;  deriv[7 * NPOS + pos] = D.cs3;
  deriv[8 * NPOS + pos] = D.s13;  deriv[9 * NPOS + pos] = D.s42;
  deriv[10 * NPOS + pos] = D.g13; deriv[11 * NPOS + pos] = D.g42;
  deriv[12 * NPOS + pos] = D.x0;  deriv[13 * NPOS + pos] = D.y0;
}

// K6: gather-fold lb / count, 1x1 conv, BN(1ch), silu -> out (9216 f32).
__global__ __launch_bounds__(1024) void final_kernel(
    const float* __restrict__ deriv, const float* __restrict__ fw,
    const float* __restrict__ gamma, const float* __restrict__ beta,
    float* __restrict__ out) {
  __shared__ float sbuf[32];
  float s = 0.f, ss = 0.f;
  for (int idx = threadIdx.x; idx < NPIX; idx += blockDim.x) {
    int y = idx / Wimg, x = idx % Wimg;
    int plo = max(0, y - (HP - 1)), phi = min(Ppat - 1, y);
    int klo = max(0, x - (WP - 1)), khi = min(Ppat - 1, x);
    float acc = 0.f;
    for (int p = plo; p <= phi; ++p) {
      int h = y - p;
      float gy = grid8(p);
      for (int k = klo; k <= khi; ++k) {
        int w = x - k;
        int pos = h * WP + w;
        Deriv D;
        D.sn0 = deriv[0 * NPOS + pos];  D.cs0 = deriv[1 * NPOS + pos];
        D.sn1 = deriv[2 * NPOS + pos];  D.cs1 = deriv[3 * NPOS + pos];
        D.sn2 = deriv[4 * NPOS + pos];  D.cs2 = deriv[5 * NPOS + pos];
        D.sn3 = deriv[6 * NPOS + pos];  D.cs3 = deriv[7 * NPOS + pos];
        D.s13 = deriv[8 * NPOS + pos];  D.s42 = deriv[9 * NPOS + pos];
        D.g13 = deriv[10 * NPOS + pos]; D.g42 = deriv[11 * NPOS + pos];
        D.x0  = deriv[12 * NPOS + pos]; D.y0  = deriv[13 * NPOS + pos];
        float lx = grid8(k) - D.x0;
        float ly = gy - D.y0;
        float d13, d42;
        dist_pix(D, lx, ly, d13, d42);
        float mad = (d13 < 0.f) ? -d13
                  : ((d42 < 0.f) ? fminf(d13, -d42) : fminf(d13, d42));
        float t = mad * INV_DELTA;
        acc += 1.f / (1.f + t * t);
      }
    }
    float cnt = (float)((phi - plo + 1) * (khi - klo + 1));  // fold(ones)
    float v = (acc / cnt) * fw[0];
    out[idx] = v;
    s += v; ss += v * v;
  }
  float sum = block_reduce_sum(s, sbuf);
  float sqs = block_reduce_sum(ss, sbuf);
  float mean = sum / (float)NPIX;
  float var = sqs / (float)NPIX - mean * mean;
  float inv = rsqrtf(var + 1e-5f);
  float g = gamma[0], b = beta[0];
  for (int idx = threadIdx.x; idx < NPIX; idx += blockDim.x) {
    float v = out[idx];  // each thread re-reads only indices it wrote
    float bn = g * (v - mean) * inv + b;
    out[idx] = bn / (1.f + expf(-bn));  // silu
  }
}

extern "C" void kernel_launch(void* const* d_in, const int* in_sizes, int n_in,
                              void* d_out, int out_size, void* d_ws,
                              size_t ws_size, hipStream_t stream) {
  (void)in_sizes; (void)n_in; (void)out_size; (void)ws_size;
  const float* x        = (const float*)d_in[0];  // (1,16,96,96)
  const float* reduce_w = (const float*)d_in[1];  // (3,16)
  const float* rgamma   = (const float*)d_in[2];  // (3,)
  const float* rbeta    = (const float*)d_in[3];  // (3,)
  const float* final_w  = (const float*)d_in[4];  // (1,1)
  const float* fgamma   = (const float*)d_in[5];  // (1,)
  const float* fbeta    = (const float*)d_in[6];  // (1,)
  float* out = (float*)d_out;                     // (1,1,96,96) f32
  float* ws = (float*)d_ws;

  float* xr     = ws;                  // 3*9216  = 27648 floats
  float* params = xr + 3 * NPIX;       // 5*7921  = 39605 floats
  float* deriv  = params + 5 * NPOS;   // 14*7921 = 110894 floats (~713 KB total)

  // K1: channel reduction via f32 WMMA (576 full waves).
  reduce_wmma_kernel<<<72, 256, 0, stream>>>(x, reduce_w, xr);
  // K2: BN + silu per channel.
  bn_silu3_kernel<<<1, 1024, 0, stream>>>(xr, rgamma, rbeta);
  // K3: params start at zero every call.
  zero_kernel<<<(5 * NPOS + 255) / 256, 256, 0, stream>>>(params, 5 * NPOS);
  // K4: 8 serialized coordinate-descent rounds.
  for (int r = 0; r < 8; ++r)
    descent_kernel<<<(NPOS + 7) / 8, 256, 0, stream>>>(xr, params, r);
  // K5: derived geometry for final gather.
  derive_kernel<<<(NPOS + 255) / 256, 256, 0, stream>>>(params, deriv);
  // K6: fold + 1x1 conv + BN + silu.
  final_kernel<<<1, 1024, 0, stream>>>(deriv, final_w, fgamma, fbeta, out);
}